// AffinityCosineLoss_43946105372782
// MI455X (gfx1250) — compile-verified
//
#include <hip/hip_runtime.h>
#include <hip/hip_bf16.h>

typedef __attribute__((ext_vector_type(2))) float v2f;
typedef __attribute__((ext_vector_type(4))) float v4f;
typedef __attribute__((ext_vector_type(8))) float v8f;

#define B_ROWS 4096
#define D_FULL 256
#define D_USE  192          // int(256 * 0.75)
#define L_CLS  128
#define TILE   128          // macro tile edge
#define NTB    (B_ROWS / TILE)      // 32 tile rows/cols
#define NTRI   (NTB * (NTB + 1) / 2) // 528 upper-tri macro tiles
#define KCHUNK 16
#define LDS_STRIDE 20       // 128-row panel, padded stride (bank-conflict free, 16B aligned)

// ---------------------------------------------------------------------------
// Kernel 1: per-row inverse norms of z = y_pred[:, :192]; one wave per row.
// Also zeroes the double loss accumulator (block 0, thread 0).
// ---------------------------------------------------------------------------
__global__ __launch_bounds__(256)
void invnorm_kernel(const float* __restrict__ y_pred,
                    float* __restrict__ inv_norm,
                    double* __restrict__ loss_acc) {
    if (blockIdx.x == 0 && threadIdx.x == 0) *loss_acc = 0.0;
    int wave = threadIdx.x >> 5;
    int lane = threadIdx.x & 31;
    int row  = blockIdx.x * 8 + wave;      // 512 blocks * 8 waves = 4096 rows
    const float* zr = y_pred + (size_t)row * D_FULL;
    float ss = 0.0f;
    #pragma unroll
    for (int c = 0; c < D_USE / 32; ++c) {
        float v = zr[lane + c * 32];
        ss += v * v;
    }
    #pragma unroll
    for (int m = 16; m; m >>= 1) ss += __shfl_xor(ss, m, 32);
    if (lane == 0) inv_norm[row] = 1.0f / fmaxf(sqrtf(ss), 1e-8f);
}

// ---------------------------------------------------------------------------
// Kernel 2: fused Gram-matrix WMMA GEMM + target + |.| + reduction.
// 256 threads (8 wave32) per block; block owns one 128x128 macro tile of the
// upper triangle. Wave w = (wr = w%4, wc = w/4) owns a 32x64 region =
// 2x4 sub-tiles of 16x16 (v_wmma_f32_16x16x4_f32 accumulators).
// ---------------------------------------------------------------------------
__global__ __launch_bounds__(256)
void gram_loss_kernel(const float* __restrict__ y_pred,
                      const int*   __restrict__ y_true,
                      const float* __restrict__ lookup,
                      const float* __restrict__ inv_norm,
                      double* __restrict__ loss_acc) {
    __shared__ float Apan[TILE * LDS_STRIDE];
    __shared__ float Bpan[TILE * LDS_STRIDE];

    // decode blockIdx -> upper-triangular tile (tr, tc), tc >= tr
    int bid = blockIdx.x;
    int tr = 0;
    while (bid >= NTB - tr) { bid -= NTB - tr; ++tr; }
    int tc = tr + bid;
    const int rb = tr * TILE;
    const int cb = tc * TILE;

    const int tid  = threadIdx.x;
    const int wave = tid >> 5;
    const int lane = tid & 31;
    const int wr   = wave & 3;     // 0..3 -> 32-row strip
    const int wc   = wave >> 2;    // 0..1 -> 64-col strip
    const int half = lane >> 4;    // lane group (0: lanes 0-15, 1: lanes 16-31)
    const int l16  = lane & 15;

    v8f acc[2][4];
    const v8f vzero = {0.f, 0.f, 0.f, 0.f, 0.f, 0.f, 0.f, 0.f};
    #pragma unroll
    for (int s = 0; s < 2; ++s)
        #pragma unroll
        for (int t = 0; t < 4; ++t) acc[s][t] = vzero;

    // cooperative panel-load mapping: 128 rows x 16 k-floats per panel,
    // 256 threads -> each thread moves 8 floats (two float4) per panel.
    const int prow = tid >> 1;
    const int pcol = (tid & 1) * 8;

    for (int k0 = 0; k0 < D_USE; k0 += KCHUNK) {
        const float* gA = y_pred + (size_t)(rb + prow) * D_FULL + k0 + pcol;
        const float* gB = y_pred + (size_t)(cb + prow) * D_FULL + k0 + pcol;
        *(v4f*)&Apan[prow * LDS_STRIDE + pcol]     = *(const v4f*)(gA);
        *(v4f*)&Apan[prow * LDS_STRIDE + pcol + 4] = *(const v4f*)(gA + 4);
        *(v4f*)&Bpan[prow * LDS_STRIDE + pcol]     = *(const v4f*)(gB);
        *(v4f*)&Bpan[prow * LDS_STRIDE + pcol + 4] = *(const v4f*)(gB + 4);
        __syncthreads();

        #pragma unroll
        for (int kk = 0; kk < KCHUNK; kk += 4) {
            // A frag (16x4 f32): lane holds row M=l16, K = kk + 2*half + {0,1}
            v2f af[2], bf[4];
            #pragma unroll
            for (int s = 0; s < 2; ++s) {
                int r = wr * 32 + s * 16 + l16;
                af[s] = *(const v2f*)&Apan[r * LDS_STRIDE + kk + 2 * half];
            }
            #pragma unroll
            for (int t = 0; t < 4; ++t) {
                int c = wc * 64 + t * 16 + l16;
                bf[t] = *(const v2f*)&Bpan[c * LDS_STRIDE + kk + 2 * half];
            }
            #pragma unroll
            for (int s = 0; s < 2; ++s)
                #pragma unroll
                for (int t = 0; t < 4; ++t)
                    acc[s][t] = __builtin_amdgcn_wmma_f32_16x16x4_f32(
                        false, af[s], false, bf[t], (short)0, acc[s][t],
                        false, false);
        }
        __syncthreads();
    }

    // ---- fused epilogue: target construction + |latent - target| + reduce ----
    float lsum = 0.0f;
    #pragma unroll
    for (int t = 0; t < 4; ++t) {
        const int   gcol = cb + wc * 64 + t * 16 + l16;   // N = lane%16
        const int   tj   = y_true[gcol];
        const float invj = inv_norm[gcol];
        #pragma unroll
        for (int s = 0; s < 2; ++s) {
            #pragma unroll
            for (int v = 0; v < 8; ++v) {
                const int grow = rb + wr * 32 + s * 16 + half * 8 + v; // M = v + 8*half
                if (gcol > grow) {     // strict upper triangle only
                    const int   ti   = y_true[grow];
                    const float invi = inv_norm[grow];
                    const float latent = 0.5f * (acc[s][t][v] * invi * invj + 1.0f);
                    const bool bi = (ti == -1), bj = (tj == -1);
                    float target;
                    if (bi && bj)       target = 0.2f;   // BG_SIM
                    else if (bi != bj)  target = 0.01f;  // BG_OTHER_SIM
                    else {
                        const bool valid = ((unsigned)ti < (unsigned)L_CLS) &&
                                           ((unsigned)tj < (unsigned)L_CLS);
                        const int ci = min(max(ti, 0), L_CLS - 1);
                        const int cj = min(max(tj, 0), L_CLS - 1);
                        target = valid ? lookup[ci * L_CLS + cj] : 0.0f;
                    }
                    lsum += fabsf(latent - target);
                }
            }
        }
    }
    #pragma unroll
    for (int m = 16; m; m >>= 1) lsum += __shfl_xor(lsum, m, 32);
    if (lane == 0) atomicAdd(loss_acc, (double)lsum);   // global_atomic_add_f64
}

// ---------------------------------------------------------------------------
// Kernel 3: finalize scalar
// ---------------------------------------------------------------------------
__global__ void finalize_kernel(const double* __restrict__ loss_acc,
                                float* __restrict__ out) {
    if (threadIdx.x == 0 && blockIdx.x == 0) {
        const double n_pairs = (double)B_ROWS * (double)(B_ROWS - 1) * 0.5;
        out[0] = (float)(loss_acc[0] / n_pairs);
    }
}

// ---------------------------------------------------------------------------
extern "C" void kernel_launch(void* const* d_in, const int* in_sizes, int n_in,
                              void* d_out, int out_size, void* d_ws, size_t ws_size,
                              hipStream_t stream) {
    const int*   y_true = (const int*)d_in[0];    // (4096,) int32
    const float* y_pred = (const float*)d_in[1];  // (4096, 256) f32
    const float* lookup = (const float*)d_in[2];  // (128, 128) f32

    // workspace layout: [0..7] double accumulator, [64..] 4096 f32 inv norms
    double* loss_acc = (double*)d_ws;
    float*  inv_norm = (float*)((char*)d_ws + 64);

    invnorm_kernel<<<B_ROWS / 8, 256, 0, stream>>>(y_pred, inv_norm, loss_acc);
    gram_loss_kernel<<<NTRI, 256, 0, stream>>>(y_pred, y_true, lookup,
                                               inv_norm, loss_acc);
    finalize_kernel<<<1, 32, 0, stream>>>(loss_acc, (float*)d_out);
}